// WeightedSumGCN_78116865179890
// MI455X (gfx1250) — compile-verified
//
#include <hip/hip_runtime.h>
#include <hip/hip_bf16.h>
#include <stdint.h>

typedef __attribute__((ext_vector_type(2))) float v2f;
typedef __attribute__((ext_vector_type(8))) float v8f;

#define DN 128  // D_IN == D_OUT == 128

// ---------------- kernel 1: softmax over 4 logits ----------------
__global__ void k_softmax4(const float* __restrict__ logits, float* __restrict__ w4) {
    if (threadIdx.x == 0 && blockIdx.x == 0) {
        float l0 = logits[0], l1 = logits[1], l2 = logits[2], l3 = logits[3];
        float m = fmaxf(fmaxf(l0, l1), fmaxf(l2, l3));
        float e0 = expf(l0 - m), e1 = expf(l1 - m), e2 = expf(l2 - m), e3 = expf(l3 - m);
        float inv = 1.0f / (e0 + e1 + e2 + e3);
        w4[0] = e0 * inv; w4[1] = e1 * inv; w4[2] = e2 * inv; w4[3] = e3 * inv;
    }
}

// ---------------- kernel 2: deg init (self-loop contributes 1.0 to deg[col=i]) ----
__global__ void k_deg_init(float* __restrict__ deg, int n) {
    int i = blockIdx.x * blockDim.x + threadIdx.x;
    if (i < n) deg[i] = 1.0f;
}

// ---------------- kernel 3: edge weight gather + degree scatter ----------------
// ew[e] = sum_k w[k] * A[k, r, c]  — only gather at edge positions, never build A_meta.
__global__ void k_edge_ew(const float* __restrict__ A, const long long* __restrict__ ei,
                          const float* __restrict__ w4, float* __restrict__ ew,
                          float* __restrict__ deg, int E, int n) {
    int e = blockIdx.x * blockDim.x + threadIdx.x;
    if (e >= E) return;
    long long r = ei[e];
    long long c = ei[(long long)E + e];
    size_t NN = (size_t)n * (size_t)n;
    size_t base = (size_t)r * (size_t)n + (size_t)c;
    float v = w4[0] * A[base]
            + w4[1] * A[base + NN]
            + w4[2] * A[base + 2 * NN]
            + w4[3] * A[base + 3 * NN];
    ew[e] = v;
    atomicAdd(&deg[c], v);
}

// ---------------- kernel 4: dinv = deg>0 ? rsqrt(max(deg,1e-30)) : 0 ----------------
__global__ void k_dinv(const float* __restrict__ deg, float* __restrict__ dinv, int n) {
    int i = blockIdx.x * blockDim.x + threadIdx.x;
    if (i < n) {
        float d = deg[i];
        dinv[i] = (d > 0.0f) ? rsqrtf(fmaxf(d, 1e-30f)) : 0.0f;
    }
}

// ---------------- kernel 5: xw = x @ W^T via V_WMMA_F32_16X16X4_F32 ----------------
// One wave per 16x16 output tile; K=128 in 32 steps of 4.
// A frag (16x4 f32): lanes 0-15 hold (M=lane, K=0..1), lanes 16-31 hold (M=lane-16, K=2..3).
// B frag (4x16 f32): lanes 0-15 hold (K=0..1, N=lane), lanes 16-31 hold (K=2..3, N=lane-16).
// C/D (16x16 f32, 8 VGPRs): vgpr j, lanes 0-15 -> (M=j, N=lane); lanes 16-31 -> (M=j+8, N=lane-16).
__global__ __launch_bounds__(256) void k_gemm_wmma(const float* __restrict__ x,
                                                   const float* __restrict__ W,
                                                   float* __restrict__ xw, int n) {
    const int wave = threadIdx.x >> 5;
    const int lane = threadIdx.x & 31;
    const int tile = blockIdx.x * 8 + wave;        // (n/16) * (128/16) tiles
    const int tilesN = DN / 16;                    // 8
    const int tileM = tile / tilesN;
    const int tileN = tile % tilesN;
    const int rowBase = tileM * 16;
    const int nBase = tileN * 16;

    const int lm = lane & 15;
    const int khalf = (lane >> 4) * 2;             // 0 or 2

    const float* __restrict__ xrow = x + (size_t)(rowBase + lm) * DN;
    const float* __restrict__ wrow = W + (size_t)(nBase + lm) * DN;  // B[k][nBase+lm] = W[(nBase+lm)*DN + k]

    v8f acc = {};
#pragma unroll
    for (int k = 0; k < DN; k += 4) {
        v2f a = *(const v2f*)(xrow + k + khalf);   // 8-byte aligned
        v2f bfrag = *(const v2f*)(wrow + k + khalf);
        acc = __builtin_amdgcn_wmma_f32_16x16x4_f32(
            /*neg_a=*/false, a, /*neg_b=*/false, bfrag,
            /*c_mod=*/(short)0, acc, /*reuse_a=*/false, /*reuse_b=*/false);
    }

    const int m0 = rowBase + (lane >> 4) * 8;
    const int col = nBase + lm;
#pragma unroll
    for (int j = 0; j < 8; ++j) {
        xw[(size_t)(m0 + j) * DN + col] = acc[j];
    }
}

// ---------------- kernel 6: out = b + dinv[i]^2 * xw[i]  (self-loop + bias, inits out) ----
__global__ void k_self_bias(const float* __restrict__ xw, const float* __restrict__ dinv,
                            const float* __restrict__ b, float* __restrict__ out, int n) {
    int i = blockIdx.x;       // row
    int d = threadIdx.x;      // 0..127
    float s = dinv[i];
    out[(size_t)i * DN + d] = b[d] + s * s * xw[(size_t)i * DN + d];
}

// ---------------- kernel 7: edge scatter: out[c] += dinv[r]*ew*dinv[c] * xw[r] ----------
__global__ __launch_bounds__(256) void k_edge_scatter(const float* __restrict__ xw,
                                                      const float* __restrict__ dinv,
                                                      const float* __restrict__ ew,
                                                      const long long* __restrict__ ei,
                                                      float* __restrict__ out, int E) {
    const int wave = threadIdx.x >> 5;
    const int lane = threadIdx.x & 31;
    const int e = blockIdx.x * 8 + wave;
    if (e >= E) return;
    long long r = ei[e];
    long long c = ei[(long long)E + e];
    float norm = dinv[r] * ew[e] * dinv[c];
    const float4* __restrict__ src = (const float4*)(xw + (size_t)r * DN);
    float4 v = src[lane];                       // 32 lanes * 4 = 128 values
    float* dst = out + (size_t)c * DN + lane * 4;
    atomicAdd(dst + 0, norm * v.x);
    atomicAdd(dst + 1, norm * v.y);
    atomicAdd(dst + 2, norm * v.z);
    atomicAdd(dst + 3, norm * v.w);
}

extern "C" void kernel_launch(void* const* d_in, const int* in_sizes, int n_in,
                              void* d_out, int out_size, void* d_ws, size_t ws_size,
                              hipStream_t stream) {
    const float*     x      = (const float*)d_in[0];
    const float*     A_list = (const float*)d_in[1];
    const long long* ei     = (const long long*)d_in[2];
    const float*     wlog   = (const float*)d_in[3];
    const float*     W      = (const float*)d_in[4];
    const float*     b      = (const float*)d_in[5];
    float* out = (float*)d_out;

    const int n = in_sizes[0] / DN;      // 8192
    const int E = in_sizes[2] / 2;       // 262144

    // workspace layout (all offsets multiples of 256B)
    char* ws = (char*)d_ws;
    float* w4   = (float*)(ws);                               // 4 floats (padded to 256)
    float* ewv  = (float*)(ws + 256);                         // E floats
    float* deg  = (float*)(ws + 256 + (size_t)E * 4);         // n floats
    float* dinv = (float*)(ws + 256 + (size_t)E * 4 + (size_t)n * 4);
    float* xw   = (float*)(ws + 256 + (size_t)E * 4 + (size_t)n * 8);

    k_softmax4<<<1, 32, 0, stream>>>(wlog, w4);
    k_deg_init<<<(n + 255) / 256, 256, 0, stream>>>(deg, n);
    k_edge_ew<<<(E + 255) / 256, 256, 0, stream>>>(A_list, ei, w4, ewv, deg, E, n);
    k_dinv<<<(n + 255) / 256, 256, 0, stream>>>(deg, dinv, n);

    const int tiles = (n / 16) * (DN / 16);                   // 4096 tiles, 8 waves/block
    k_gemm_wmma<<<tiles / 8, 256, 0, stream>>>(x, W, xw, n);

    k_self_bias<<<n, DN, 0, stream>>>(xw, dinv, b, out, n);
    k_edge_scatter<<<(E + 7) / 8, 256, 0, stream>>>(xw, dinv, ewv, ei, out, E);
}